// GradientRefinementModule_70446053589299
// MI455X (gfx1250) — compile-verified
//
#include <hip/hip_runtime.h>

// GradientRefinementModule for MI455X (gfx1250, wave32).
// Latency-bound scattered-gather iteration: one lane per (row, tracked column),
// lane pairs exchange their step via a pure-VALU DPP8 xor-1 permute.

#define BATCH      16384
#define SIG_LEN    8192
#define BASE_STEP  0.002f
#define MAX_ITERS  50
#define MIN_DELTA  0.0001f
#define DSTEP      0.01f
#define IDX_SCALE  8191.0f   // (SIG_LEN - 1)

// DPP8 selector [1,0,3,2,5,4,7,6]: swap adjacent lanes (xor 1) within each
// group of 8. Lane pairs (2k, 2k+1) never straddle a group-of-8 boundary.
#define DPP8_XOR1  0xDE54C1

__device__ __forceinline__ float clamp01(float x) {
    return fminf(fmaxf(x, 0.0f), 1.0f);
}

__device__ __forceinline__ int clampi(int v) {
    return min(max(v, 0), SIG_LEN - 1);
}

// Pair-swap via v_mov_b32_dpp8 (VALU only — no LDS/DScnt round trip).
__device__ __forceinline__ float pair_swap(float x) {
    int xi = __builtin_bit_cast(int, x);
    int yi = __builtin_amdgcn_mov_dpp8(xi, DPP8_XOR1);
    return __builtin_bit_cast(float, yi);
}

__global__ __launch_bounds__(256) void grad_refine_kernel(
        const float* __restrict__ signal,
        const float* __restrict__ peaks,
        float* __restrict__ out) {
    const int tid  = blockIdx.x * blockDim.x + threadIdx.x;
    const int row  = tid >> 1;
    const int side = tid & 1;                // 0 -> column 0, 1 -> column 2
    if (row >= BATCH) return;

    const float* __restrict__ sig = signal + (size_t)row * SIG_LEN;

    float p    = peaks[row * 3 + (side ? 2 : 0)];
    float mid  = peaks[row * 3 + 1];
    bool  conv      = false;
    bool  convOther = false;

    // Warm L1/L2 around the starting neighborhood (gfx1250 global_prefetch).
    // All iterations sample within a slowly drifting +/-82-element window.
    {
        int ci = clampi((int)(clamp01(p) * IDX_SCALE));
        __builtin_prefetch(&sig[clampi(ci - 96)], 0, 0);
        __builtin_prefetch(&sig[ci],              0, 0);
        __builtin_prefetch(&sig[clampi(ci + 96)], 0, 0);
    }

    for (int it = 0; it < MAX_ITERS; ++it) {
        const float pc = p;
        const float pl = clamp01(p - DSTEP);
        const float pr = clamp01(p + DSTEP);

        // --- compute all six gather addresses first so the loads overlap ---
        const float ic = pc * IDX_SCALE;
        const float il = pl * IDX_SCALE;
        const float ir = pr * IDX_SCALE;

        const int c0 = clampi((int)floorf(ic));
        const int c1 = clampi((int)ceilf(ic));
        const int l0 = clampi((int)floorf(il));
        const int l1 = clampi((int)ceilf(il));
        const int r0 = clampi((int)floorf(ir));
        const int r1 = clampi((int)ceilf(ir));

        // six independent global_load_b32 issued back-to-back (s_clause)
        const float vc0 = sig[c0];
        const float vc1 = sig[c1];
        const float vl0 = sig[l0];
        const float vl1 = sig[l1];
        const float vr0 = sig[r0];
        const float vr1 = sig[r1];

        const float wc = ic - (float)c0;
        const float wl = il - (float)l0;
        const float wr = ir - (float)r0;

        const float v  = vc0 * (1.0f - wc) + vc1 * wc;
        const float vL = vl0 * (1.0f - wl) + vl1 * wl;
        const float vR = vr0 * (1.0f - wr) + vr1 * wr;

        const float g = (vR - vL) / (2.0f * DSTEP);
        const float c = (vR - 2.0f * v + vL) / (DSTEP * DSTEP);

        float u = (c < 0.0f) ? (-BASE_STEP * g) : 0.0f;
        if (conv) u = 0.0f;                          // post-zeroing, as reference
        conv = conv || (fabsf(u) < MIN_DELTA);

        // exchange step with the partner lane (other tracked column) — VALU DPP8
        const float uOther = pair_swap(u);
        convOther = convOther || (fabsf(uOther) < MIN_DELTA);

        p   = clamp01(p + u);
        mid = clamp01(mid + 0.5f * (u + uOther));    // commutative: both lanes agree

        // Once both columns are converged, upd==0 forever -> positions frozen.
        if (conv && convOther) break;
    }

    // Final ordering swap. pos0 lives on even lane, pos2 on odd lane.
    const float pOther = pair_swap(p);
    const float pos0 = side ? pOther : p;
    const float pos2 = side ? p : pOther;
    const bool ordered = pos0 < pos2;

    if (side == 0) {
        out[row * 3 + 0] = ordered ? pos0 : pos2;
        out[row * 3 + 1] = ordered ? mid  : 0.5f * (pos0 + pos2);
    } else {
        out[row * 3 + 2] = ordered ? pos2 : pos0;
    }
}

extern "C" void kernel_launch(void* const* d_in, const int* in_sizes, int n_in,
                              void* d_out, int out_size, void* d_ws, size_t ws_size,
                              hipStream_t stream) {
    const float* signal = (const float*)d_in[0];   // (16384, 8192) f32
    const float* peaks  = (const float*)d_in[1];   // (16384, 3)    f32
    float* out          = (float*)d_out;           // (16384, 3)    f32

    const int threads = BATCH * 2;                 // one lane per (row, column)
    dim3 block(256);                               // 8 wave32 per block
    dim3 grid((threads + block.x - 1) / block.x);
    grad_refine_kernel<<<grid, block, 0, stream>>>(signal, peaks, out);
}